// IM2HT_10247791968804
// MI455X (gfx1250) — compile-verified
//
#include <hip/hip_runtime.h>
#include <stdint.h>

// Problem constants (fixed by the reference).
constexpr int IMP        = 128 * 128;      // pixels per channel = 16384
constexpr int BC         = 2 * 32;         // flattened (b,c) channels = 64
constexpr int HT_H       = 183;
constexpr int HT_W       = 60;
constexpr int N_THETA    = 60;
constexpr int HT         = HT_H * HT_W;    // 10980
constexpr int OUT_N      = BC * HT;        // 702720
constexpr int VPT        = IMP;            // votes per theta = 16384
constexpr int N_SEG      = 8;
constexpr int SEG_VOTES  = VPT / N_SEG;    // 2048
constexpr int CHUNK      = 512;            // votes staged per async chunk
constexpr int N_CHUNK    = SEG_VOTES / CHUNK;  // 4
constexpr int CHUNK_B    = CHUNK * 12;     // 6144 bytes (3 f32 per vote)
constexpr int ACC_N      = HT_H * BC;      // 11712 LDS accumulator floats

__global__ void ht_zero_kernel(float* __restrict__ out, int n) {
    int i = blockIdx.x * blockDim.x + threadIdx.x;
    if (i < n) out[i] = 0.0f;
}

// im (64 x 16384) -> imT (16384 x 64), 32x32 LDS tiles, fully coalesced both ways.
__global__ void ht_xpose_kernel(const float* __restrict__ im, float* __restrict__ imT) {
    __shared__ float tile[32][33];
    const int p0 = blockIdx.x * 32;     // pixel tile
    const int c0 = blockIdx.y * 32;     // channel tile
    const int tx = threadIdx.x, ty = threadIdx.y;
    for (int j = ty; j < 32; j += 8)
        tile[j][tx] = im[(size_t)(c0 + j) * IMP + (p0 + tx)];
    __syncthreads();
    for (int j = ty; j < 32; j += 8)
        imT[(size_t)(p0 + j) * BC + (c0 + tx)] = tile[tx][j];
}

template <bool XP>
__global__ __launch_bounds__(256) void ht_vote_kernel(const float* __restrict__ vm,
                                                      const float* __restrict__ img,
                                                      float* __restrict__ out) {
    __shared__ float acc[ACC_N];              // 46848 B: [rho][bc] bins for this theta
    __shared__ float stage[2][CHUNK * 3];     // 12288 B: double-buffered vote rows

    const int tid   = threadIdx.x;
    const int theta = blockIdx.x;
    const int seg   = blockIdx.y;
    const int lane  = tid & 31;
    const int wv    = tid >> 5;               // 8 waves
    const int bc    = ((wv & 1) << 5) | lane; // channel handled by this lane
    const int vslot = wv >> 2;                // unused; kept simple below
    (void)vslot;
    const int vgrp  = wv >> 1;                // 4 vote slots, 2 waves (64 ch) each

    for (int i = tid; i < ACC_N; i += 256) acc[i] = 0.0f;

    const char* gsrc0 = (const char*)vm + (size_t)(theta * VPT + seg * SEG_VOTES) * 12;
    const unsigned lds0 = (unsigned)(uintptr_t)&stage[0][0];   // low 32 bits = LDS offset

    auto issue_chunk = [&](int c, int buf) {
        const char*    g = gsrc0 + (size_t)c * CHUNK_B;
        const unsigned l = lds0 + (unsigned)buf * CHUNK_B;
        // 6144 B / 16 B = 384 lane-transfers; 256 lanes -> 2 rounds (EXEC-masked).
        for (int ofs = tid * 16; ofs < CHUNK_B; ofs += 256 * 16) {
            asm volatile("global_load_async_to_lds_b128 %0, %1, off"
                         :: "v"(l + (unsigned)ofs), "v"(g + ofs) : "memory");
        }
    };

    issue_chunk(0, 0);
    for (int c = 0; c < N_CHUNK; ++c) {
        const int buf = c & 1;
        asm volatile("s_wait_asynccnt 0x0" ::: "memory");  // chunk c landed in LDS
        __syncthreads();                                   // visible to all waves
        if (c + 1 < N_CHUNK) issue_chunk(c + 1, (c + 1) & 1);

        const float* sb = &stage[buf][0];
        // Each vote-slot (pair of waves) owns 128 contiguous votes of the chunk.
        for (int g = 0; g < 4; ++g) {
            const int vl = vgrp * (CHUNK / 4) + g * 32 + lane;  // per-lane vote
            // Stride-3 DWORD gather: conflict-free (gcd(3,64)==1).
            const float pf = sb[vl * 3 + 0];
            const float hf = sb[vl * 3 + 1];
            const float wf = sb[vl * 3 + 2];
            const int pixOff = XP ? ((int)pf) * BC : (int)pf;
            const int accOff = (((int)hf) / HT_W) * BC;         // rho*64
            const int wbits  = __float_as_int(wf);
#pragma unroll
            for (int k = 0; k < 32; ++k) {
                const int   po = __builtin_amdgcn_readlane(pixOff, k);
                const int   ao = __builtin_amdgcn_readlane(accOff, k);
                const float w  = __int_as_float(__builtin_amdgcn_readlane(wbits, k));
                const float v  = XP ? img[po + bc] : img[(size_t)bc * IMP + po];
                atomicAdd(&acc[ao + bc], v * w);                // ds_add_f32, no conflicts
            }
        }
        __syncthreads();   // all reads of this buffer done before it is re-filled
    }

    // Flush bins: theta columns are disjoint, but 8 segments share them -> global atomics.
    for (int i = tid; i < ACC_N; i += 256) {
        const int rho = i >> 6;
        const int ch  = i & 63;
        atomicAdd(&out[(size_t)ch * HT + rho * HT_W + theta], acc[i]);
    }
}

extern "C" void kernel_launch(void* const* d_in, const int* in_sizes, int n_in,
                              void* d_out, int out_size, void* d_ws, size_t ws_size,
                              hipStream_t stream) {
    const float* im  = (const float*)d_in[0];
    const float* vm  = (const float*)d_in[1];
    float*       out = (float*)d_out;
    (void)in_sizes; (void)n_in; (void)out_size;

    ht_zero_kernel<<<(OUT_N + 255) / 256, 256, 0, stream>>>(out, OUT_N);

    const bool xp = ws_size >= (size_t)IMP * BC * sizeof(float);
    if (xp) {
        float* imT = (float*)d_ws;
        ht_xpose_kernel<<<dim3(IMP / 32, BC / 32), dim3(32, 8), 0, stream>>>(im, imT);
        ht_vote_kernel<true><<<dim3(N_THETA, N_SEG), 256, 0, stream>>>(vm, imT, out);
    } else {
        ht_vote_kernel<false><<<dim3(N_THETA, N_SEG), 256, 0, stream>>>(vm, im, out);
    }
}